// S4DKernel_6828998001033
// MI455X (gfx1250) — compile-verified
//
#include <hip/hip_runtime.h>

typedef __attribute__((ext_vector_type(2))) float v2f;
typedef __attribute__((ext_vector_type(8))) float v8f;

#define S4D_H 1024
#define S4D_N 64
#define S4D_L 4096
#define PAD 65   // odd stride -> bank-conflict-free row and column access

// One block per head h. 256 threads = 8 wave32.
// Phase 1: build U (64x64, A-operand, [l0][n]) and W (64x64, B-operand, [n][l1]) in LDS.
// Phase 2: K[h] = 2*(Ure*Wre + (-Uim)*Wim) via v_wmma_f32_16x16x4_f32.
__global__ __launch_bounds__(256) void s4d_wmma_kernel(
    const float* __restrict__ C_re, const float* __restrict__ C_im,
    const float* __restrict__ log_dt, const float* __restrict__ log_A_real,
    const float* __restrict__ A_imag, float* __restrict__ out)
{
  __shared__ float sUre [64 * PAD];
  __shared__ float sUimN[64 * PAD];   // holds -Uim
  __shared__ float sWre [64 * PAD];
  __shared__ float sWim [64 * PAD];

  const int h = blockIdx.x;
  const int t = threadIdx.x;
  const int n = t & 63;     // state index this thread owns
  const int q = t >> 6;     // quarter of the l-range (0..3)

  // ---- per-(h,n) scalars -------------------------------------------------
  const float dt = __expf(log_dt[h]);
  const float Ar = -__expf(log_A_real[h * S4D_N + n]);
  const float Ai = A_imag[h * S4D_N + n];
  const float dr = Ar * dt;           // Re(dtA)  (negative -> decaying exp)
  const float di = Ai * dt;           // Im(dtA)

  // E-1 = exp(dtA) - 1
  float er = __expf(dr);
  float sn, cs;
  __sincosf(di, &sn, &cs);
  const float Er = er * cs - 1.0f;
  const float Ei = er * sn;
  // Q = (E-1)/A = (E-1)*conj(A)/|A|^2
  const float inv = 1.0f / (Ar * Ar + Ai * Ai);
  const float Qr = (Er * Ar + Ei * Ai) * inv;
  const float Qi = (Ei * Ar - Er * Ai) * inv;
  // C_final = C * Q
  const float Cr = C_re[h * S4D_N + n];
  const float Ci = C_im[h * S4D_N + n];
  const float Fr = Cr * Qr - Ci * Qi;
  const float Fi = Cr * Qi + Ci * Qr;

  // ---- fill LDS tiles ----------------------------------------------------
  #pragma unroll
  for (int i = 0; i < 16; ++i) {
    const int   l  = q * 16 + i;
    const float lf = (float)l;

    // W[n][l1] = exp(dtA * l1)
    float e1 = __expf(dr * lf);
    float s1, c1;
    __sincosf(di * lf, &s1, &c1);
    sWre[n * PAD + l] = e1 * c1;
    sWim[n * PAD + l] = e1 * s1;

    // U[l0][n] = C_final * exp(dtA * 64*l0)   (store -Im part)
    const float lf64 = lf * 64.0f;
    float e2 = __expf(dr * lf64);
    float s2, c2;
    __sincosf(di * lf64, &s2, &c2);
    const float vr = e2 * c2, vi = e2 * s2;
    sUre [l * PAD + n] = Fr * vr - Fi * vi;
    sUimN[l * PAD + n] = -(Fr * vi + Fi * vr);
  }

  __syncthreads();

  // ---- WMMA phase: 16 output tiles (4x4 grid of 16x16), 2 per wave -------
  const int wave  = t >> 5;
  const int lane  = t & 31;
  const int row   = lane & 15;          // A-row / B-col / D-col within tile
  const int kb    = (lane >> 4) << 1;   // K sub-offset: lanes16-31 hold K+2,K+3
  const int mhalf = (lane >> 4) << 3;   // D rows: lanes16-31 hold M+8..M+15

  float* o = out + (size_t)h * S4D_L;

  #pragma unroll
  for (int tile = 0; tile < 2; ++tile) {
    const int tid16 = wave + tile * 8;        // 0..15
    const int m0 = (tid16 >> 2) * 16;         // l0 tile origin
    const int n0 = (tid16 & 3)  * 16;         // l1 tile origin

    v8f acc = {};
    #pragma unroll
    for (int k = 0; k < 64; k += 4) {
      // A fragments (16x4 f32): VGPR0 = {K=0 | K=2}, VGPR1 = {K=1 | K=3}
      const float* uR = &sUre [(m0 + row) * PAD + k + kb];
      const float* uI = &sUimN[(m0 + row) * PAD + k + kb];
      v2f aR, aI;
      aR.x = uR[0]; aR.y = uR[1];
      aI.x = uI[0]; aI.y = uI[1];
      // B fragments (4x16 f32): VGPR0 = row {K=0 | K=2}, VGPR1 = row {K=1 | K=3}
      const float* wR = &sWre[(k + kb) * PAD + n0 + row];
      const float* wI = &sWim[(k + kb) * PAD + n0 + row];
      v2f bR, bI;
      bR.x = wR[0]; bR.y = wR[PAD];
      bI.x = wI[0]; bI.y = wI[PAD];

      // acc += Ure*Wre ; acc += (-Uim)*Wim
      acc = __builtin_amdgcn_wmma_f32_16x16x4_f32(
          false, aR, false, bR, (short)0, acc, false, false);
      acc = __builtin_amdgcn_wmma_f32_16x16x4_f32(
          false, aI, false, bI, (short)0, acc, false, false);
    }

    // D layout: VGPR v -> M = m0 + mhalf + v, N = n0 + (lane&15)
    #pragma unroll
    for (int v = 0; v < 8; ++v) {
      const int m = m0 + mhalf + v;   // global l0
      const int c = n0 + row;         // global l1
      o[m * 64 + c] = 2.0f * acc[v];
    }
  }
}

extern "C" void kernel_launch(void* const* d_in, const int* in_sizes, int n_in,
                              void* d_out, int out_size, void* d_ws, size_t ws_size,
                              hipStream_t stream) {
  (void)in_sizes; (void)n_in; (void)out_size; (void)d_ws; (void)ws_size;
  const float* C_re       = (const float*)d_in[0];
  const float* C_im       = (const float*)d_in[1];
  const float* log_dt     = (const float*)d_in[2];
  const float* log_A_real = (const float*)d_in[3];
  const float* A_imag     = (const float*)d_in[4];
  // d_in[5] is L (scalar int) == 4096, baked in at compile time.
  float* out = (float*)d_out;

  s4d_wmma_kernel<<<dim3(S4D_H), dim3(256), 0, stream>>>(
      C_re, C_im, log_dt, log_A_real, A_imag, out);
}